// DoReMiLoss_51754355917079
// MI455X (gfx1250) — compile-verified
//
#include <hip/hip_runtime.h>
#include <hip/hip_bf16.h>
#include <math.h>

// Problem constants (match reference)
#define SEQ      2048
#define BATCH    8
#define VOCAB    16000
#define NDOM     8
#define STEP_SZ  1.0f
#define SMOOTH   1e-3f

#define TPB      256          // 8 waves (wave32)
#define NWAVES   (TPB / 32)

// Tiling: each thread owns one float4 column, strided by the block.
#define VEC      4
#define CHUNK    (TPB * VEC)                 // 1024 floats per block-iteration
#define FULL     (VOCAB / CHUNK)             // 15 full iterations
#define NITER    (FULL + 1)                  // 16 register slots per thread
#define TAILV    ((VOCAB - FULL * CHUNK) / VEC)  // 160 lanes carry a tail vec4
#define CHUNK_B  (CHUNK * 4)                 // 4096 bytes per iteration stride

// -------- async global -> LDS copy (CDNA5 gfx1250) --------
// INST_OFFSET is applied to BOTH the LDS and the global address, so a single
// (lds_addr, global_addr) register pair + immediate offsets covers the row.
template <int OFF>
__device__ __forceinline__ void async_b128(unsigned lds_off, unsigned long long g) {
  asm volatile("global_load_async_to_lds_b128 %0, %1, off offset:%2"
               :
               : "v"(lds_off), "v"(g), "i"(OFF)
               : "memory");
}

template <int K>
struct CopyUnroll {
  __device__ __forceinline__ static void run(unsigned lds_off, unsigned long long g) {
    CopyUnroll<K - 1>::run(lds_off, g);
    async_b128<(K - 1) * CHUNK_B>(lds_off, g);
  }
};
template <>
struct CopyUnroll<0> {
  __device__ __forceinline__ static void run(unsigned, unsigned long long) {}
};

__device__ __forceinline__ void wait_async_zero() {
  asm volatile("s_wait_asynccnt 0" ::: "memory");
}

// -------------------------------------------------------------------
// Kernel 1: per-row exact logsumexp.
//  - async-DMA the 16000-float row into LDS (written once)
//  - single LDS read pass into 16 float4 registers per thread
//  - pass 1 (max) and pass 2 (sum exp) both run from registers
// grid.x = SEQ*BATCH, row r = s*BATCH + b  (logits are [S,B,V] row-major).
// -------------------------------------------------------------------
__global__ __launch_bounds__(TPB) void rowloss_kernel(
    const float* __restrict__ logits,             // [S,B,V]
    const int* __restrict__ label_ids,            // [B,S]
    const unsigned char* __restrict__ label_mask, // [B,S] (bool)
    float* __restrict__ tokloss)                  // [B,S]
{
  __shared__ __align__(16) float tile[VOCAB];     // 64000 B
  __shared__ float red[NWAVES];

  const int row = blockIdx.x;           // s*BATCH + b
  const int s   = row / BATCH;
  const int b   = row - s * BATCH;
  const float* __restrict__ rowp = logits + (size_t)row * VOCAB;

  const int tid  = threadIdx.x;
  const int lane = tid & 31;
  const int wid  = tid >> 5;

  // ---- async DMA: one address pair, 15 immediate-offset b128 loads + tail ----
  const unsigned lds_off = (unsigned)(size_t)(&tile[0]) + (unsigned)tid * 16u;
  const unsigned long long gaddr = (unsigned long long)(size_t)(rowp + tid * VEC);
  CopyUnroll<FULL>::run(lds_off, gaddr);
  if (tid < TAILV) async_b128<FULL * CHUNK_B>(lds_off, gaddr);
  wait_async_zero();
  __syncthreads();

  // ---- single LDS read pass into registers ----
  float4 vals[NITER];
  const float4* __restrict__ lp = (const float4*)(tile) + tid;  // stride TPB float4s
  #pragma unroll
  for (int k = 0; k < FULL; ++k) vals[k] = lp[k * TPB];
  const float ninf = -__builtin_inff();
  vals[FULL] = (tid < TAILV) ? lp[FULL * TPB] : make_float4(ninf, ninf, ninf, ninf);

  // ---- pass 1: row max (registers -> wave shuffle -> LDS tree) ----
  float m = ninf;
  #pragma unroll
  for (int k = 0; k < NITER; ++k) {
    m = fmaxf(m, fmaxf(fmaxf(vals[k].x, vals[k].y), fmaxf(vals[k].z, vals[k].w)));
  }
  #pragma unroll
  for (int o = 16; o > 0; o >>= 1) m = fmaxf(m, __shfl_xor(m, o, 32));
  if (lane == 0) red[wid] = m;
  __syncthreads();
  float rowmax = ninf;
  #pragma unroll
  for (int w = 0; w < NWAVES; ++w) rowmax = fmaxf(rowmax, red[w]);
  __syncthreads();   // red[] about to be reused

  // ---- pass 2: sum of exp(x - max) from registers (tail -INF -> exp = 0) ----
  float ssum = 0.0f;
  #pragma unroll
  for (int k = 0; k < NITER; ++k) {
    ssum += __expf(vals[k].x - rowmax) + __expf(vals[k].y - rowmax) +
            __expf(vals[k].z - rowmax) + __expf(vals[k].w - rowmax);
  }
  #pragma unroll
  for (int o = 16; o > 0; o >>= 1) ssum += __shfl_xor(ssum, o, 32);
  if (lane == 0) red[wid] = ssum;
  __syncthreads();

  if (tid == 0) {
    float total = 0.0f;
    #pragma unroll
    for (int w = 0; w < NWAVES; ++w) total += red[w];
    const int   lab = label_ids[b * SEQ + s];
    const float xl  = tile[lab];
    const float lse = rowmax + logf(total);
    const float mk  = label_mask[b * SEQ + s] ? 1.0f : 0.0f;
    tokloss[b * SEQ + s] = (lse - xl) * mk;
  }
}

// -------------------------------------------------------------------
// Kernel 2: per-sample means + DoReMi domain-weight update.
// One block, 8 waves: wave w reduces sample b=w over SEQ, then thread 0
// does the tiny 8-domain scalar tail and writes 24 floats:
//   out[0:8)   = losses
//   out[8:16)  = normalized_domain_losses
//   out[16:24) = smooth_domain_weights
// -------------------------------------------------------------------
__global__ __launch_bounds__(TPB) void finalize_kernel(
    const float* __restrict__ tokloss,            // [B,S]
    const unsigned char* __restrict__ label_mask, // [B,S]
    const int* __restrict__ domain_idxs,          // [B]
    const float* __restrict__ ref_losses,         // [B]
    const float* __restrict__ domain_weights,     // [NDOM]
    float* __restrict__ out)                      // [24]
{
  __shared__ float lsum[BATCH];
  __shared__ float lcnt[BATCH];

  const int tid  = threadIdx.x;
  const int lane = tid & 31;
  const int w    = tid >> 5;

  if (w < BATCH) {
    float sm = 0.0f, cm = 0.0f;
    for (int sIdx = lane; sIdx < SEQ; sIdx += 32) {
      sm += tokloss[w * SEQ + sIdx];
      cm += label_mask[w * SEQ + sIdx] ? 1.0f : 0.0f;
    }
    #pragma unroll
    for (int o = 16; o > 0; o >>= 1) {
      sm += __shfl_xor(sm, o, 32);
      cm += __shfl_xor(cm, o, 32);
    }
    if (lane == 0) { lsum[w] = sm; lcnt[w] = cm; }
  }
  __syncthreads();

  if (tid == 0) {
    float losses[BATCH], excess[BATCH];
    #pragma unroll
    for (int bb = 0; bb < BATCH; ++bb) {
      losses[bb] = lsum[bb] / lcnt[bb];
      float e = losses[bb] - ref_losses[bb];
      excess[bb] = e > 0.0f ? e : 0.0f;
    }
    float dl[NDOM], cnt[NDOM];
    #pragma unroll
    for (int d = 0; d < NDOM; ++d) { dl[d] = 0.0f; cnt[d] = 0.0f; }
    for (int bb = 0; bb < BATCH; ++bb) {
      int d = domain_idxs[bb];
      if (d >= 0 && d < NDOM) { dl[d] += excess[bb]; cnt[d] += 1.0f; }
    }
    float norm[NDOM], upd[NDOM], usum = 0.0f;
    #pragma unroll
    for (int d = 0; d < NDOM; ++d) {
      norm[d] = dl[d] / cnt[d];
      upd[d]  = domain_weights[d] * expf(STEP_SZ * norm[d]);
      usum   += upd[d];
    }
    const float uni = 1.0f / (float)NDOM;
    #pragma unroll
    for (int bb = 0; bb < BATCH; ++bb) out[bb] = losses[bb];
    #pragma unroll
    for (int d = 0; d < NDOM; ++d)   out[8 + d]  = norm[d];
    #pragma unroll
    for (int d = 0; d < NDOM; ++d)   out[16 + d] = (1.0f - SMOOTH) * upd[d] / usum + SMOOTH * uni;
  }
}

extern "C" void kernel_launch(void* const* d_in, const int* in_sizes, int n_in,
                              void* d_out, int out_size, void* d_ws, size_t ws_size,
                              hipStream_t stream) {
  (void)in_sizes; (void)n_in; (void)out_size; (void)ws_size;

  const float*         logits         = (const float*)d_in[0];         // [S,B,V]
  const int*           label_ids      = (const int*)d_in[1];           // [B,S]
  const unsigned char* label_mask     = (const unsigned char*)d_in[2]; // [B,S] bool
  const int*           domain_idxs    = (const int*)d_in[3];           // [B]
  const float*         ref_losses     = (const float*)d_in[4];         // [B]
  const float*         domain_weights = (const float*)d_in[5];         // [NDOM]
  float*               out            = (float*)d_out;                 // [24]

  float* tokloss = (float*)d_ws;  // [BATCH*SEQ] floats = 64 KB scratch

  rowloss_kernel<<<SEQ * BATCH, TPB, 0, stream>>>(logits, label_ids, label_mask, tokloss);
  finalize_kernel<<<1, TPB, 0, stream>>>(tokloss, label_mask, domain_idxs,
                                         ref_losses, domain_weights, out);
}